// VesselIntensitySynth_74156905333429
// MI455X (gfx1250) — compile-verified
//
#include <hip/hip_runtime.h>

// -----------------------------------------------------------------------------
// VesselIntensitySynth — label-volume flip + dual-LUT gather + one-hot expand.
//
// Memory-bound: ~335 MB of HBM traffic, ~14 us floor at 23.3 TB/s.
// CDNA5 usage: async global->LDS staging of the 4 KB LUTs (ASYNCcnt path,
// via inline asm global_load_async_to_lds_b64), s_wait_asynccnt, LDS-bank
// gathers, and non-temporal B128 streaming loads/stores so the write-once
// 256 MB output does not thrash L2.
// -----------------------------------------------------------------------------

#define VOLN   (256 * 256 * 256)   // 16,777,216 voxels
#define N_IDS  512

// Fixed axis-flip draw: jax.random.bernoulli(jax.random.key(1), 0.5, (3,)).
// Compile-time constants (patched to the actual fixed draw); flipping only
// remaps read indices, all accesses stay 16B-aligned and contiguous.
#define FLIP_D 1
#define FLIP_H 0
#define FLIP_W 1

typedef int   v4i __attribute__((ext_vector_type(4)));
typedef float v4f __attribute__((ext_vector_type(4)));

__global__ __launch_bounds__(256, 8)
void VesselIntensitySynth_74156905333429_kernel(const int*   __restrict__ labels,
                                                const float* __restrict__ ilut,
                                                const int*   __restrict__ clut,
                                                float*       __restrict__ out)
{
    __shared__ float lds_f[N_IDS];   // intensity LUT (2 KB)
    __shared__ int   lds_c[N_IDS];   // class LUT     (2 KB)

    const int t = threadIdx.x;       // 256 threads = 8 waves (wave32)

    // ---- Stage both LUTs into LDS via async global->LDS DMA -------------
    // 256 lanes x b64 = 2 KB per LUT: one async op per lane per LUT.
    // LDS dest operand is a 32-bit LDS byte offset in a VGPR; global source
    // is a 64-bit address in a VGPR pair; "off" = no SADDR.
#if defined(__gfx1250__)
    {
        const unsigned lF = (unsigned)(unsigned long long)
            (__attribute__((address_space(3))) char*)lds_f + (unsigned)(t * 8);
        const unsigned lC = (unsigned)(unsigned long long)
            (__attribute__((address_space(3))) char*)lds_c + (unsigned)(t * 8);
        const char* gF = (const char*)ilut + t * 8;
        const char* gC = (const char*)clut + t * 8;

        asm volatile("global_load_async_to_lds_b64 %0, %1, off"
                     :: "v"(lF), "v"(gF) : "memory");
        asm volatile("global_load_async_to_lds_b64 %0, %1, off"
                     :: "v"(lC), "v"(gC) : "memory");

#if __has_builtin(__builtin_amdgcn_s_wait_asynccnt)
        __builtin_amdgcn_s_wait_asynccnt(0);
#else
        asm volatile("s_wait_asynccnt 0" ::: "memory");
#endif
    }
#else
    // Portable fallback: two 32-bit LDS stores per lane per LUT.
    lds_f[t]       = ilut[t];
    lds_f[t + 256] = ilut[t + 256];
    lds_c[t]       = clut[t];
    lds_c[t + 256] = clut[t + 256];
#endif
    __syncthreads();   // every wave drained its own ASYNCcnt before arriving

    // ---- Streamed gather: 4 voxels (one b128) per thread per iteration --
    const int nquads = VOLN / 4;                       // 4,194,304
    const int stride = (int)(gridDim.x * blockDim.x);
    float* __restrict__ oh = out + VOLN;               // one-hot base [3,D,H,W]

    for (int q = (int)(blockIdx.x * blockDim.x) + t; q < nquads; q += stride) {
        const int o = q << 2;                 // output linear voxel index
        const int w = o & 255;
        const int h = (o >> 8) & 255;
        const int d = o >> 16;

        const int sd  = FLIP_D ? 255 - d : d;
        const int sh  = FLIP_H ? 255 - h : h;
        const int swb = FLIP_W ? 252 - w : w;          // aligned 4-wide base
        const int src = (((sd << 8) | sh) << 8) | swb;

        // Non-temporal b128 label load (streamed, never re-read).
        const v4i lab = __builtin_nontemporal_load((const v4i*)(labels + src));

        int l0, l1, l2, l3;
        if (FLIP_W) { l0 = lab.w; l1 = lab.z; l2 = lab.y; l3 = lab.x; }
        else        { l0 = lab.x; l1 = lab.y; l2 = lab.z; l3 = lab.w; }

        // LDS gathers: 64 banks handle the random per-lane indices.
        v4f sc;
        sc.x = lds_f[l0]; sc.y = lds_f[l1]; sc.z = lds_f[l2]; sc.w = lds_f[l3];
        const int c0 = lds_c[l0], c1 = lds_c[l1], c2 = lds_c[l2], c3 = lds_c[l3];

        __builtin_nontemporal_store(sc, (v4f*)(out + o));

#pragma unroll
        for (int c = 0; c < 3; ++c) {
            v4f v;
            v.x = (c0 == c) ? 1.0f : 0.0f;
            v.y = (c1 == c) ? 1.0f : 0.0f;
            v.z = (c2 == c) ? 1.0f : 0.0f;
            v.w = (c3 == c) ? 1.0f : 0.0f;
            __builtin_nontemporal_store(v, (v4f*)(oh + c * VOLN + o));
        }
    }
}

extern "C" void kernel_launch(void* const* d_in, const int* in_sizes, int n_in,
                              void* d_out, int out_size, void* d_ws, size_t ws_size,
                              hipStream_t stream) {
    (void)in_sizes; (void)n_in; (void)out_size; (void)d_ws; (void)ws_size;

    const int*   labels = (const int*)  d_in[0];   // [256,256,256] int32
    const float* ilut   = (const float*)d_in[1];   // [512] float32
    const int*   clut   = (const int*)  d_in[2];   // [512] int32
    float*       out    = (float*)      d_out;     // [1+3, 256,256,256] float32

    // 4096 blocks x 256 threads = 1M threads -> 4 quads each (grid-stride);
    // amortizes the per-block 4 KB LUT staging while keeping 32K waves in
    // flight to saturate 23.3 TB/s.
    dim3 grid(4096), block(256);
    hipLaunchKernelGGL(VesselIntensitySynth_74156905333429_kernel,
                       grid, block, 0, stream, labels, ilut, clut, out);
}